// ERuleLoss_19877108646587
// MI455X (gfx1250) — compile-verified
//
#include <hip/hip_runtime.h>

typedef __attribute__((ext_vector_type(2))) float v2f;
typedef __attribute__((ext_vector_type(8))) float v8f;

#define NN 1000
#define LDS_STRIDE 1002  // even (8B-aligned rows), odd in 8B units -> conflict-free ds_load_b64

__global__ __launch_bounds__(256) void eruleloss_wmma_f32(
    const float* __restrict__ y_pred,
    const float* __restrict__ P,
    const float* __restrict__ M,
    float* __restrict__ out)
{
    __shared__ float ldsY[16 * LDS_STRIDE];
    __shared__ float ldsLoss[16];

    const int tid  = threadIdx.x;
    const int lane = tid & 31;
    const int wave = tid >> 5;
    const int half = lane >> 4;      // 0: lanes 0-15, 1: lanes 16-31
    const int l16  = lane & 15;
    const int rowBase = blockIdx.x * 16;

    // Stage the 16 x 1000 y_pred stripe into LDS (padded row stride).
    for (int idx = tid; idx < 16 * NN; idx += 256) {
        int r = idx / NN;
        int c = idx - r * NN;
        ldsY[r * LDS_STRIDE + c] = y_pred[(size_t)(rowBase + r) * NN + c];
    }
    if (tid < 16) ldsLoss[tid] = 0.0f;
    __syncthreads();

    const int NT = (NN + 15) / 16;   // 63 column tiles

    // Per-lane loss partials: slot r always maps to row m = 8*half + r,
    // so we accumulate across all n-tiles and reduce once at the end.
    float lsum[8] = {0.f, 0.f, 0.f, 0.f, 0.f, 0.f, 0.f, 0.f};

    // A fragment base: row m = l16, K offset selects this half's K pair.
    const float* aBase = &ldsY[l16 * LDS_STRIDE + half * 2];

    for (int j = wave; j < NT; j += 8) {
        const int  n      = j * 16 + l16;          // output column this lane owns
        const bool nvalid = (n < NN);
        const int  nc     = nvalid ? n : (NN - 1); // clamp for safe addressing
        const float invM  = nvalid ? (1.0f / M[nc]) : 0.0f; // zero kills tail cols
        const float* bBase = P + (size_t)nc * NN + half * 2;

        v8f acc = {};

        // Exactly 2x unroll: pairs the two A-fragment LDS reads into one
        // ds_load_2addr_b64 and clause-groups the two P loads, without
        // triggering full unroll (which spills 500 VGPRs of A to scratch).
        #pragma clang loop unroll_count(2)
        for (int k0 = 0; k0 < NN; k0 += 4) {
            v2f a = *(const v2f*)(aBase + k0);       // y_pred[m, k0+off .. +1]
            v2f p = *(const v2f*)(bBase + k0);       // P[n, k0+off .. +1]
            v2f b;
            b.x = p.x * invM;                        // (P/M)^T fragment
            b.y = p.y * invM;
            // D = A(16x4, f32) x B(4x16, f32) + C ; exact fp32 semantics.
            acc = __builtin_amdgcn_wmma_f32_16x16x4_f32(
                /*neg_a=*/false, a, /*neg_b=*/false, b,
                /*c_mod=*/(short)0, acc, /*reuse_a=*/false, /*reuse_b=*/false);
        }

        // Fold this tile into per-lane partials. For tail columns acc is
        // exactly zero (B was scaled by invM == 0), so reading the
        // clamped-index weight unconditionally is safe and branch-free.
        const int m0 = half * 8;
        #pragma unroll
        for (int r = 0; r < 8; ++r) {
            float w = ldsY[(m0 + r) * LDS_STRIDE + nc];
            lsum[r] += acc[r] * w;
        }
    }

    // One cross-lane reduction per wave (16 lanes per half share a row m).
    #pragma unroll
    for (int r = 0; r < 8; ++r) {
        float v = lsum[r];
        v += __shfl_xor(v, 1, 16);
        v += __shfl_xor(v, 2, 16);
        v += __shfl_xor(v, 4, 16);
        v += __shfl_xor(v, 8, 16);
        if (l16 == 0) atomicAdd(&ldsLoss[half * 8 + r], v);  // ds_add_f32
    }

    __syncthreads();
    if (tid < 16) {
        out[rowBase + tid] = ldsLoss[tid] * (1.0f / (float)NN);
    }
}

extern "C" void kernel_launch(void* const* d_in, const int* in_sizes, int n_in,
                              void* d_out, int out_size, void* d_ws, size_t ws_size,
                              hipStream_t stream) {
    const float* y_pred = (const float*)d_in[0];
    // d_in[1] = y_true, unused by the loss
    const float* P = (const float*)d_in[2];
    const float* M = (const float*)d_in[3];
    float* out = (float*)d_out;

    const int B = in_sizes[0] / NN;       // 16384
    const int grid = B / 16;              // 1024 workgroups, 16 rows each
    eruleloss_wmma_f32<<<grid, 256, 0, stream>>>(y_pred, P, M, out);
}